// HouseholderFrameBuilder_71640054497898
// MI455X (gfx1250) — compile-verified
//
#include <hip/hip_runtime.h>
#include <hip/hip_bf16.h>

// CDNA5 / gfx1250, wave32.
typedef __attribute__((ext_vector_type(2))) float v2f;
typedef __attribute__((ext_vector_type(8))) float v8f;

#define HF_K 4

__device__ __forceinline__ v8f wmma4_f32(v2f a, v2f b, v8f c) {
  // V_WMMA_F32_16X16X4_F32 : D(16x16,f32) = A(16x4,f32) * B(4x16,f32) + C
  return __builtin_amdgcn_wmma_f32_16x16x4_f32(
      false, a, false, b, (short)0, c, false, false);
}

// Raw 32-bit LDS address of a generic pointer that points into shared memory
// (flat->LDS mapping uses addr[31:0] per CDNA5 flat addressing rules).
__device__ __forceinline__ unsigned lds_addr_of(const void* p) {
  return (unsigned)(uintptr_t)(__attribute__((address_space(3))) const void*)p;
}

// GLOBAL_LOAD_ASYNC_TO_LDS_B128 (GV mode): DMA 16B from global memory
// straight into LDS, tracked by ASYNCcnt. vdst = LDS byte address VGPR.
__device__ __forceinline__ void async_load_to_lds_b128(unsigned lds_byte_addr,
                                                       const void* gaddr) {
  asm volatile("global_load_async_to_lds_b128 %0, %1, off"
               :: "v"(lds_byte_addr), "v"(gaddr)
               : "memory");
}

__device__ __forceinline__ void wait_asynccnt0() {
#if __has_builtin(__builtin_amdgcn_s_wait_asynccnt)
  __builtin_amdgcn_s_wait_asynccnt(0);
#else
  asm volatile("s_wait_asynccnt 0" ::: "memory");
#endif
}

// ---------------------------------------------------------------------------
// Kernel 1: frames + Householder composition.  One 64-thread block per node.
// Thread t owns row t of U (64 fp32 in registers).
// ---------------------------------------------------------------------------
__global__ void __launch_bounds__(64)
hh_frames_kernel(const float* __restrict__ h,
                 const float* __restrict__ W1, const float* __restrict__ b1,
                 const float* __restrict__ W2, const float* __restrict__ b2,
                 float* __restrict__ U) {
  const int node = blockIdx.x;
  const int t = threadIdx.x;  // 0..63

  __shared__ float sh[64];     // input features
  __shared__ float shid[128];  // hidden (post-SiLU)
  __shared__ float sfv[256];   // frame vectors, sfv[i*64 + j]

  sh[t] = h[(size_t)node * 64 + t];
  __syncthreads();

  // hidden = silu(h @ W1 + b1), thread t computes units t and t+64
  float a0 = b1[t], a1 = b1[t + 64];
  #pragma unroll 8
  for (int d = 0; d < 64; ++d) {
    const float hd = sh[d];
    a0 = fmaf(hd, W1[d * 128 + t], a0);
    a1 = fmaf(hd, W1[d * 128 + t + 64], a1);
  }
  a0 = a0 / (1.0f + __expf(-a0));
  a1 = a1 / (1.0f + __expf(-a1));
  shid[t] = a0;
  shid[t + 64] = a1;
  __syncthreads();

  // frame = hidden @ W2 + b2, thread t computes units t, t+64, t+128, t+192
  float f0 = b2[t], f1 = b2[t + 64], f2 = b2[t + 128], f3 = b2[t + 192];
  #pragma unroll 8
  for (int d = 0; d < 128; ++d) {
    const float hd = shid[d];
    f0 = fmaf(hd, W2[d * 256 + t], f0);
    f1 = fmaf(hd, W2[d * 256 + t + 64], f1);
    f2 = fmaf(hd, W2[d * 256 + t + 128], f2);
    f3 = fmaf(hd, W2[d * 256 + t + 192], f3);
  }
  sfv[t] = f0;
  sfv[64 + t] = f1;
  sfv[128 + t] = f2;
  sfv[192 + t] = f3;
  __syncthreads();

  // Householder composition: U row t in registers.
  float Urow[64];
  #pragma unroll
  for (int j = 0; j < 64; ++j) Urow[j] = (j == t) ? 1.0f : 0.0f;

  #pragma unroll
  for (int i = 0; i < HF_K; ++i) {
    float uv = 0.0f, nrm = 0.0f;
    #pragma unroll
    for (int j = 0; j < 64; ++j) {
      const float vj = sfv[i * 64 + j];  // LDS broadcast
      uv = fmaf(Urow[j], vj, uv);
      nrm = fmaf(vj, vj, nrm);
    }
    nrm = fmaxf(nrm, 1e-6f);
    const float s = 2.0f * uv / nrm;
    #pragma unroll
    for (int j = 0; j < 64; ++j) Urow[j] = fmaf(-s, sfv[i * 64 + j], Urow[j]);
  }

  float4* dst = (float4*)(U + (size_t)node * 4096 + (size_t)t * 64);
  #pragma unroll
  for (int q = 0; q < 16; ++q)
    dst[q] = make_float4(Urow[4 * q], Urow[4 * q + 1],
                         Urow[4 * q + 2], Urow[4 * q + 3]);
}

// ---------------------------------------------------------------------------
// Kernel 2: Q[e] = U[col]^T @ U[row] via V_WMMA_F32_16X16X4_F32.
// One 256-thread block (8 waves) per edge. U tiles DMA'd into LDS with
// GLOBAL_LOAD_ASYNC_TO_LDS_B128 (source is L2-resident: 163 MB U fits the
// 192 MB L2). Q streamed out with non-temporal stores.
// ---------------------------------------------------------------------------
__global__ void __launch_bounds__(256)
hh_edge_q_kernel(const float* __restrict__ U, const int* __restrict__ ei,
                 float* __restrict__ Q, int E) {
  const int e = blockIdx.x;
  const int t = threadIdx.x;   // 0..255
  const int lane = t & 31;
  const int wave = t >> 5;

  __shared__ float lds[8192];
  float* Uc = lds;          // U[col]  (16 KB)
  float* Ur = lds + 4096;   // U[row]  (16 KB)

  const int row = ei[e];
  const int col = ei[E + e];

  // Async DMA staging: 2 x 16 KB, each thread issues 8 x B128 transfers.
  const unsigned lds_base = lds_addr_of(lds);
  const char* gc = (const char*)(U + (size_t)col * 4096);
  const char* gr = (const char*)(U + (size_t)row * 4096);
  #pragma unroll
  for (int q = 0; q < 4; ++q) {
    const unsigned boff = (unsigned)((t + q * 256) * 16);
    async_load_to_lds_b128(lds_base + boff,          gc + boff);
    async_load_to_lds_b128(lds_base + 16384u + boff, gr + boff);
  }
  wait_asynccnt0();
  __syncthreads();

  // Tile assignment: 16 output tiles (4x4 grid of 16x16); wave w does
  // i0 = (w&3)*16 and the pair j0, j0+16 with j0 = (w>>2)*32, sharing A.
  const int i0 = (wave & 3) * 16;
  const int j0 = (wave >> 2) * 32;

  // WMMA f32 A(16x4)/B(4x16) lane layout (ISA 7.12.2):
  //   lanes 0-15 : element m(or n) = lane,    VGPR0 -> K=0, VGPR1 -> K=1
  //   lanes 16-31: element m(or n) = lane-16, VGPR0 -> K=2, VGPR1 -> K=3
  const int m = lane & 15;
  const int kk = (lane >> 4) * 2;

  v8f acc0 = {};
  v8f acc1 = {};

  #pragma unroll
  for (int kb = 0; kb < 64; kb += 4) {
    const int k0 = kb + kk;
    v2f a, b0, b1;
    // A[mrow, k] = Uc[k, i0+mrow]   (Q = Uc^T * Ur)
    a.x = Uc[k0 * 64 + i0 + m];
    a.y = Uc[(k0 + 1) * 64 + i0 + m];
    // B[k, ncol] = Ur[k, j0+ncol]
    b0.x = Ur[k0 * 64 + j0 + m];
    b0.y = Ur[(k0 + 1) * 64 + j0 + m];
    b1.x = Ur[k0 * 64 + j0 + 16 + m];
    b1.y = Ur[(k0 + 1) * 64 + j0 + 16 + m];
    acc0 = wmma4_f32(a, b0, acc0);
    acc1 = wmma4_f32(a, b1, acc1);
  }

  // C/D layout: VGPR g -> M = g + (lane/16)*8, N = lane%16.
  // Non-temporal stores keep the 819 MB Q stream from evicting U out of L2.
  float* qbase = Q + (size_t)e * 4096;
  const int mhalf = (lane >> 4) << 3;
  #pragma unroll
  for (int g = 0; g < 8; ++g) {
    const int mr = i0 + g + mhalf;
    __builtin_nontemporal_store(acc0[g], qbase + (size_t)mr * 64 + j0 + m);
    __builtin_nontemporal_store(acc1[g], qbase + (size_t)mr * 64 + j0 + 16 + m);
  }
}

// ---------------------------------------------------------------------------
// Launch: inputs in setup_inputs() order: h, edge_index, W1, b1, W2, b2.
// d_out = [U (N*4096) | Q (E*4096)] fp32.
// ---------------------------------------------------------------------------
extern "C" void kernel_launch(void* const* d_in, const int* in_sizes, int n_in,
                              void* d_out, int out_size, void* d_ws, size_t ws_size,
                              hipStream_t stream) {
  const float* h  = (const float*)d_in[0];
  const int*   ei = (const int*)d_in[1];
  const float* W1 = (const float*)d_in[2];
  const float* b1 = (const float*)d_in[3];
  const float* W2 = (const float*)d_in[4];
  const float* b2 = (const float*)d_in[5];

  const int N = in_sizes[0] / 64;   // 10000
  const int E = in_sizes[1] / 2;    // 50000

  float* U = (float*)d_out;
  float* Q = (float*)d_out + (size_t)N * 4096;

  hh_frames_kernel<<<N, 64, 0, stream>>>(h, W1, b1, W2, b2, U);
  hh_edge_q_kernel<<<E, 256, 0, stream>>>(U, ei, Q, E);
}